// OnlineTripletLossMining_22814866276815
// MI455X (gfx1250) — compile-verified
//
#include <hip/hip_runtime.h>
#include <hip/hip_bf16.h>

typedef __attribute__((ext_vector_type(2))) float v2f;
typedef __attribute__((ext_vector_type(8))) float v8f;

#define NN 512
#define DD 128
#define FEPS 1e-16f

// ---------------------------------------------------------------------------
// Kernel 1: G = X * X^T via V_WMMA_F32_16X16X4_F32 (full fp32 precision).
// One wave (32 lanes) computes one 16x16 tile of G; K-loop of 32 WMMA ops.
//
// A-matrix 16x4 layout (ISA 7.12.2): lanes 0-15 -> M=lane, VGPR0=K0,VGPR1=K1;
// lanes 16-31 -> M=lane-16, VGPR0=K2,VGPR1=K3.
// B-matrix 4x16: rows striped across lanes within a VGPR: lanes 0-15 -> K=0/1,
// lanes 16-31 -> K=2/3, column N=lane&15.  For B = X^T tile this is the SAME
// per-lane load pattern as A, just with the nt-tile row of X.
// ---------------------------------------------------------------------------
__global__ __launch_bounds__(32) void gram_wmma_f32(const float* __restrict__ X,
                                                    float* __restrict__ G) {
  const int lane = threadIdx.x;              // 0..31
  const int mt = blockIdx.x;                 // 0..31  (rows of G)
  const int nt = blockIdx.y;                 // 0..31  (cols of G)
  const int r = lane & 15;
  const int khalf = (lane >> 4) * 2;         // 0 or 2

  const float* __restrict__ arow = X + (size_t)(mt * 16 + r) * DD + khalf;
  const float* __restrict__ brow = X + (size_t)(nt * 16 + r) * DD + khalf;

  v8f acc = {};
#pragma unroll 8
  for (int k = 0; k < DD; k += 4) {
    v2f a, b;
    a.x = arow[k];
    a.y = arow[k + 1];
    b.x = brow[k];
    b.y = brow[k + 1];
    // (neg_a, A, neg_b, B, c_mod, C, reuse_a, reuse_b)
    acc = __builtin_amdgcn_wmma_f32_16x16x4_f32(false, a, false, b,
                                                (short)0, acc, false, false);
  }

  // C/D 16x16 f32 layout: VGPR v, lane l: M = v + (l>=16 ? 8 : 0), N = l&15.
  const int rowbase = mt * 16 + ((lane >> 4) * 8);
  const int col = nt * 16 + (lane & 15);
#pragma unroll
  for (int v = 0; v < 8; ++v) {
    G[(size_t)(rowbase + v) * NN + col] = acc[v];
  }
}

// ---------------------------------------------------------------------------
// Kernel 2: distance matrix with the reference's exact eps/mask handling.
// d0 = relu(n_i + n_j - 2 g); mask = (d0==0); d = sqrt(d0+mask*EPS+EPS)*(1-mask)+EPS
// ---------------------------------------------------------------------------
__global__ __launch_bounds__(256) void dist_kernel(const float* __restrict__ G,
                                                   float* __restrict__ Dm) {
  const int j = blockIdx.x * blockDim.x + threadIdx.x;
  const int i = blockIdx.y;
  if (j >= NN) return;
  const float g  = G[(size_t)i * NN + j];
  const float ni = G[(size_t)i * NN + i];
  const float nj = G[(size_t)j * NN + j];
  float d0 = ni + nj - 2.0f * g;
  d0 = fmaxf(d0, 0.0f);
  const float mask = (d0 == 0.0f) ? 1.0f : 0.0f;
  const float d1 = d0 + mask * FEPS;
  const float dd = sqrtf(d1 + FEPS) * (1.0f - mask) + FEPS;
  Dm[(size_t)i * NN + j] = dd;
}

// ---------------------------------------------------------------------------
// Kernel 3: per-anchor triplet reduction. One block per anchor i.
// valid(i,j,k): label_j==label_i && j!=i  and  label_k!=label_i
// (j!=k, k!=i, k!=j follow automatically).  Distance row + labels live in LDS.
// Deterministic: fixed thread->k mapping, fixed-order LDS tree reduction,
// per-block partials (no float atomics).
// ---------------------------------------------------------------------------
__global__ __launch_bounds__(256) void triplet_kernel(const float* __restrict__ Dm,
                                                      const int* __restrict__ labels,
                                                      const float* __restrict__ bias,
                                                      float* __restrict__ partial) {
  __shared__ float s_d[NN];
  __shared__ int   s_lab[NN];
  __shared__ float s_sum[256];
  __shared__ float s_cnt[256];

  const int i = blockIdx.x;
  const int t = threadIdx.x;

  for (int idx = t; idx < NN; idx += 256) {
    s_d[idx]   = Dm[(size_t)i * NN + idx];
    s_lab[idx] = labels[idx];
  }
  __syncthreads();

  const int   li = s_lab[i];
  const float b  = bias[0];

  float sum = 0.0f;
  float cnt = 0.0f;
  for (int j = 0; j < NN; ++j) {
    if (j == i || s_lab[j] != li) continue;     // positives only (~7 of 512)
    const float dij_b = s_d[j] + b;
    for (int k = t; k < NN; k += 256) {
      if (s_lab[k] == li) continue;             // negatives only
      float l = dij_b - s_d[k];
      l = fmaxf(l, 0.0f);
      sum += l;
      cnt += (l > 1e-16f) ? 1.0f : 0.0f;
    }
  }

  s_sum[t] = sum;
  s_cnt[t] = cnt;
  __syncthreads();
#pragma unroll
  for (int s = 128; s > 0; s >>= 1) {
    if (t < s) {
      s_sum[t] += s_sum[t + s];
      s_cnt[t] += s_cnt[t + s];
    }
    __syncthreads();
  }
  if (t == 0) {
    partial[i]      = s_sum[0];
    partial[NN + i] = s_cnt[0];
  }
}

// ---------------------------------------------------------------------------
// Kernel 4: deterministic final reduction of 512 partial (sum, count) pairs.
// ---------------------------------------------------------------------------
__global__ __launch_bounds__(512) void finalize_kernel(const float* __restrict__ partial,
                                                       float* __restrict__ out) {
  __shared__ float s_s[NN];
  __shared__ float s_c[NN];
  const int t = threadIdx.x;
  s_s[t] = partial[t];
  s_c[t] = partial[NN + t];
  __syncthreads();
#pragma unroll
  for (int s = 256; s > 0; s >>= 1) {
    if (t < s) {
      s_s[t] += s_s[t + s];
      s_c[t] += s_c[t + s];
    }
    __syncthreads();
  }
  if (t == 0) out[0] = s_s[0] / (s_c[0] + 1e-16f);
}

extern "C" void kernel_launch(void* const* d_in, const int* in_sizes, int n_in,
                              void* d_out, int out_size, void* d_ws, size_t ws_size,
                              hipStream_t stream) {
  const float* X      = (const float*)d_in[0];   // [512,128] fp32
  const int*   labels = (const int*)d_in[1];     // [512] int
  const float* bias   = (const float*)d_in[2];   // [1] fp32
  float* out = (float*)d_out;

  char* ws = (char*)d_ws;
  float* G       = (float*)ws;                                       // 1 MB
  float* Dm      = (float*)(ws + (size_t)NN * NN * sizeof(float));   // 1 MB
  float* partial = (float*)(ws + 2 * (size_t)NN * NN * sizeof(float)); // 4 KB

  (void)in_sizes; (void)n_in; (void)out_size; (void)ws_size;

  // 1) Gram matrix via fp32 WMMA: one wave per 16x16 tile.
  dim3 gGrid(NN / 16, NN / 16);
  gram_wmma_f32<<<gGrid, 32, 0, stream>>>(X, G);

  // 2) Distance matrix.
  dist_kernel<<<dim3(NN / 256, NN), 256, 0, stream>>>(G, Dm);

  // 3) Per-anchor triplet partial sums/counts.
  triplet_kernel<<<NN, 256, 0, stream>>>(Dm, labels, bias, partial);

  // 4) Final deterministic reduction + division.
  finalize_kernel<<<1, NN, 0, stream>>>(partial, out);
}